// DecoderAllSentinel_61091614818466
// MI455X (gfx1250) — compile-verified
//
#include <hip/hip_runtime.h>

#define B_     64
#define TMAX_  256
#define P_     64
#define D_     128
#define C_     10
#define TBASE_ 367
#define CP_    16   // C padded to WMMA tile width

typedef __attribute__((ext_vector_type(2))) float v2f;
typedef __attribute__((ext_vector_type(8))) float v8f;

// ---------------------------------------------------------------------------
// Kernel 1: ragged lengths per batch, prefix offsets, token->(bi,ti) maps,
//           and the [N,2] masks tail of the output (as floats).
// ---------------------------------------------------------------------------
__global__ void k_ragged(const int* __restrict__ dates, int* __restrict__ off,
                         int* __restrict__ bi_arr, int* __restrict__ ti_arr,
                         float* __restrict__ masks, int N)
{
  __shared__ int sL[B_];
  __shared__ int soff[B_ + 1];
  int t = threadIdx.x;
  if (t < B_) {
    int L = TMAX_ - 1;                       // matches np.argmax(all-False)==0 case
    for (int j = TMAX_ - 1; j >= 0; --j) {
      if (dates[t * TMAX_ + j] != 0) { L = j; break; }
    }
    sL[t] = L;
  }
  __syncthreads();
  if (t == 0) {
    int s = 0;
    for (int i = 0; i < B_; ++i) { soff[i] = s; s += sL[i]; }
    soff[B_] = s;
    for (int i = 0; i <= B_; ++i) off[i] = soff[i];
  }
  __syncthreads();
  for (int n = t; n < N; n += blockDim.x) {
    int lo = 0, hi = B_;                     // largest i with soff[i] <= n
    while (hi - lo > 1) { int mid = (lo + hi) >> 1; if (soff[mid] <= n) lo = mid; else hi = mid; }
    int ti = n - soff[lo];
    bi_arr[n] = lo; ti_arr[n] = ti;
    masks[2 * n]     = (float)lo;
    masks[2 * n + 1] = (float)ti;
  }
}

// ---------------------------------------------------------------------------
// Kernel 2: sinusoid table [TBASE, D] in double precision (matches float64
//           reference construction), cast to f32.
// ---------------------------------------------------------------------------
__global__ void k_table(float* __restrict__ table)
{
  int idx = blockIdx.x * blockDim.x + threadIdx.x;
  if (idx >= TBASE_ * D_) return;
  int tpos = idx / D_;
  int d = idx % D_;
  double freq = exp(-2.0 * (double)(d >> 1) / (double)D_ * log((double)TBASE_));
  double ang  = (double)tpos * freq;
  double val  = (d & 1) ? cos(ang) : sin(ang);
  table[idx] = (float)val;
}

// ---------------------------------------------------------------------------
// Kernel 3: TWb[t, c] = table[t,:] . W[c,:] + b[c]   ([TBASE, CP_], c>=C_ -> 0)
// ---------------------------------------------------------------------------
__global__ void k_twb(const float* __restrict__ table, const float* __restrict__ W,
                      const float* __restrict__ bias, float* __restrict__ TWb)
{
  int idx = blockIdx.x * blockDim.x + threadIdx.x;
  if (idx >= TBASE_ * CP_) return;
  int tpos = idx >> 4;
  int c = idx & (CP_ - 1);
  float acc = 0.f;
  if (c < C_) {
    const float* tr = table + tpos * D_;
    const float* wr = W + c * D_;
    for (int d = 0; d < D_; ++d) acc += tr[d] * wr[d];
    acc += bias[c];
  }
  TWb[idx] = acc;
}

// ---------------------------------------------------------------------------
// Kernel 4: XW[row, c] = x_flat[row,:] . W[c,:]  via V_WMMA_F32_16X16X4_F32.
//           One wave (32 lanes) per 16x16 output tile; 32 K-steps of 4.
//           A layout: lane l holds row l%16, K pair (k0, k0+1) for l<16 else
//           (k0+2, k0+3). B mirrors with N striped across lanes; rows >= C_
//           are zeroed with a lane-select (no EXEC divergence at the WMMA).
// ---------------------------------------------------------------------------
__global__ void __launch_bounds__(32) k_xw(const float* __restrict__ x,
                                           const float* __restrict__ W,
                                           float* __restrict__ XW)
{
  int lane  = threadIdx.x;
  int row16 = lane & 15;
  int hi    = lane >> 4;                       // 0: K+0..1, 1: K+2..3
  int r0    = blockIdx.x * 16;                 // tile row base in [0, 4096)
  const float* xr = x + (size_t)(r0 + row16) * D_;
  const float* wr = W + (size_t)(row16 < C_ ? row16 : 0) * D_;
  float bmask = (row16 < C_) ? 1.f : 0.f;

  v8f acc = {};
#pragma unroll
  for (int k0 = 0; k0 < D_; k0 += 4) {
    v2f a  = *(const v2f*)(xr + k0 + 2 * hi);
    v2f bb = *(const v2f*)(wr + k0 + 2 * hi);
    bb *= bmask;
    acc = __builtin_amdgcn_wmma_f32_16x16x4_f32(
        /*neg_a=*/false, a, /*neg_b=*/false, bb,
        /*c_mod=*/(short)0, acc, /*reuse_a=*/false, /*reuse_b=*/false);
  }
  // C/D layout: VGPR v, lanes 0-15 -> M=v, lanes 16-31 -> M=v+8; N = lane%16
#pragma unroll
  for (int v = 0; v < 8; ++v) {
    int m = v + 8 * hi;
    XW[(size_t)(r0 + m) * CP_ + row16] = acc[v];
  }
}

// ---------------------------------------------------------------------------
// Kernel 5: streaming writer (bandwidth-bound part).
//           recons[n, c, p] = XW[bi*P_+p, c] + TWb[date, c]
//           XW batch slab staged in LDS with stride-17 pad (conflict-free).
// ---------------------------------------------------------------------------
__global__ void __launch_bounds__(256) k_write(const int* __restrict__ bi_arr,
                                               const int* __restrict__ ti_arr,
                                               const int* __restrict__ dates,
                                               const float* __restrict__ XW,
                                               const float* __restrict__ TWb,
                                               float* __restrict__ out)
{
  __shared__ float sxw[P_ * 17];
  __shared__ float stw[CP_];
  int n = blockIdx.x;
  int t = threadIdx.x;
  int bi = bi_arr[n];
  int ti = ti_arr[n];
  int date = dates[bi * TMAX_ + ti];
  const float* xw = XW + (size_t)bi * P_ * CP_;
  for (int i = t; i < P_ * CP_; i += 256) {
    int p = i >> 4, c = i & 15;
    sxw[p * 17 + c] = xw[i];
  }
  if (t < CP_) stw[t] = TWb[date * CP_ + t];
  __syncthreads();
  float* op = out + (size_t)n * (C_ * P_);
  for (int i = t; i < C_ * P_; i += 256) {   // 640 floats, coalesced
    int c = i >> 6, p = i & 63;
    op[i] = sxw[p * 17 + c] + stw[c];
  }
}

// ---------------------------------------------------------------------------
extern "C" void kernel_launch(void* const* d_in, const int* in_sizes, int n_in,
                              void* d_out, int out_size, void* d_ws, size_t ws_size,
                              hipStream_t stream)
{
  (void)in_sizes; (void)n_in; (void)ws_size;
  const float* x     = (const float*)d_in[0];
  // d_in[1] = attentions: unused by the reference output
  const int*   dates = (const int*)d_in[2];
  const float* W     = (const float*)d_in[3];
  const float* bias  = (const float*)d_in[4];
  float* out = (float*)d_out;

  int N = out_size / (C_ * P_ + 2);   // out = [N,C,P] recons + [N,2] masks

  // workspace carve (all offsets 64B-aligned)
  char* p = (char*)d_ws;
  int*   off    = (int*)(p + 0);                                   //    512 B
  int*   bi_arr = (int*)(p + 512);                                 //  65536 B
  int*   ti_arr = (int*)(p + 512 + 65536);                         //  65536 B
  float* table  = (float*)(p + 512 + 131072);                      // 187904 B
  float* TWb    = (float*)(p + 512 + 131072 + 187904);             //  23488 B
  float* XW     = (float*)(p + 512 + 131072 + 187904 + 23488);     // 262144 B

  k_ragged<<<1, 256, 0, stream>>>(dates, off, bi_arr, ti_arr,
                                  out + (size_t)N * C_ * P_, N);
  k_table<<<(TBASE_ * D_ + 255) / 256, 256, 0, stream>>>(table);
  k_twb<<<(TBASE_ * CP_ + 255) / 256, 256, 0, stream>>>(table, W, bias, TWb);
  k_xw<<<(B_ * P_) / 16, 32, 0, stream>>>(x, W, XW);
  k_write<<<N, 256, 0, stream>>>(bi_arr, ti_arr, dates, XW, TWb, out);
}